// AttentionPooling_36404142801078
// MI455X (gfx1250) — compile-verified
//
#include <hip/hip_runtime.h>

// AttentionPooling for MI455X (gfx1250, wave32).
// x:[B,T,D] f32, mask:[B,T] i32, q:[1,1,D] f32 -> pooled:[B,D] f32
// Single HBM pass over x: flash-style online softmax partials per (b, chunk),
// scores computed with V_WMMA_F32_16X16X4_F32 (16 row-dots per wave).

typedef float v2f __attribute__((ext_vector_type(2)));
typedef float v4f __attribute__((ext_vector_type(4)));
typedef float v8f __attribute__((ext_vector_type(8)));

#define BATCH   32
#define SEQT    8192
#define DIM     512
#define TC      128            // rows (t) per chunk
#define CHUNKS  (SEQT / TC)    // 64
#define TPB     256            // 8 waves
#define PITCH   516            // LDS row pitch in floats (bank-conflict padding, 16B aligned)
#define NEGBIG  (-1.0e9f)
#define PARTSTRIDE (DIM + 2)   // acc[512], m, s

__global__ __launch_bounds__(TPB) void attnpool_partial_kernel(
    const float* __restrict__ x,
    const int*   __restrict__ mask,
    const float* __restrict__ q,
    float*       __restrict__ partial)
{
    __shared__ float xs[TC * PITCH];   // 264192 B staged x tile
    __shared__ float q_lds[DIM];       // 2048 B
    __shared__ float sc_lds[TC];       // masked+scaled scores
    __shared__ float p_lds[TC];        // exp(score - m)

    const int chunk = blockIdx.x;
    const int b     = blockIdx.y;
    const int tid   = threadIdx.x;
    const int lane  = tid & 31;
    const int wv    = tid >> 5;

    const float* xrow0 = x + ((size_t)b * SEQT + (size_t)chunk * TC) * DIM;

    // ---- Stage tile: 128 rows x 512 f32, coalesced b128 loads, one HBM pass ----
    for (int i = tid; i < TC * (DIM / 4); i += TPB) {
        int r  = i >> 7;        // / (DIM/4)
        int c4 = i & 127;
        v4f v = *(const v4f*)(xrow0 + (size_t)r * DIM + c4 * 4);
        *(v4f*)(&xs[r * PITCH + c4 * 4]) = v;
    }
    for (int i = tid; i < DIM; i += TPB) q_lds[i] = q[i];
    __syncthreads();

    // ---- Phase 1: scores via V_WMMA_F32_16X16X4_F32, each wave owns 16 rows ----
    // A layout (32-bit 16x4): lanes 0-15 -> rows, K = k0 + {0,1}; lanes 16-31 -> K = k0 + {2,3}.
    // B = q broadcast across all 16 N columns: identical per-lane float2 addressing into q.
    const int rowbase = wv * 16;
    const int arow    = rowbase + (lane & 15);
    const int koff    = (lane >= 16) ? 2 : 0;
    const float* asrc = &xs[arow * PITCH + koff];
    const float* bsrc = &q_lds[koff];

    v8f c0 = {}; v8f c1 = {}; v8f c2 = {}; v8f c3 = {};
    for (int k = 0; k < DIM; k += 16) {
        v2f a0 = *(const v2f*)(asrc + k);
        v2f b0 = *(const v2f*)(bsrc + k);
        v2f a1 = *(const v2f*)(asrc + k + 4);
        v2f b1 = *(const v2f*)(bsrc + k + 4);
        v2f a2 = *(const v2f*)(asrc + k + 8);
        v2f b2 = *(const v2f*)(bsrc + k + 8);
        v2f a3 = *(const v2f*)(asrc + k + 12);
        v2f b3 = *(const v2f*)(bsrc + k + 12);
        c0 = __builtin_amdgcn_wmma_f32_16x16x4_f32(false, a0, false, b0, (short)0, c0, false, false);
        c1 = __builtin_amdgcn_wmma_f32_16x16x4_f32(false, a1, false, b1, (short)0, c1, false, false);
        c2 = __builtin_amdgcn_wmma_f32_16x16x4_f32(false, a2, false, b2, (short)0, c2, false, false);
        c3 = __builtin_amdgcn_wmma_f32_16x16x4_f32(false, a3, false, b3, (short)0, c3, false, false);
    }
    v8f c = (c0 + c1) + (c2 + c3);

    // C layout: lane 0 (N=0) holds rows rowbase+0..7 in c[0..7];
    //           lane 16 (N=0, M=8..15) holds rows rowbase+8..15.
    const float scale = 0.044194173824159216f; // 1/sqrt(512)
    if (lane == 0 || lane == 16) {
        const int rb = rowbase + ((lane == 16) ? 8 : 0);
        const int* mrow = mask + (size_t)b * SEQT + (size_t)chunk * TC;
        #pragma unroll
        for (int i = 0; i < 8; ++i) {
            float sc = c[i] * scale;
            sc += (1.0f - (float)mrow[rb + i]) * NEGBIG;
            sc_lds[rb + i] = sc;
        }
    }
    __syncthreads();

    // ---- Phase 2: chunk-local online-softmax terms ----
    float m = -3.4e38f;
    #pragma unroll 8
    for (int t = 0; t < TC; ++t) m = fmaxf(m, sc_lds[t]);   // LDS broadcast reads
    if (tid < TC) p_lds[tid] = __expf(sc_lds[tid] - m);
    __syncthreads();

    // ---- Phase 3: weighted accumulation from LDS; each thread owns 2 columns ----
    const int d0 = tid * 2;
    v2f acc = {0.0f, 0.0f};
    float s = 0.0f;
    #pragma unroll 4
    for (int t = 0; t < TC; ++t) {
        float p = p_lds[t];
        s += p;                                  // identical in every thread
        v2f xv = *(const v2f*)(&xs[t * PITCH + d0]);
        acc += p * xv;
    }

    float* pp = partial + ((size_t)b * CHUNKS + chunk) * PARTSTRIDE;
    *(v2f*)(pp + d0) = acc;
    if (tid == 0) { pp[DIM] = m; pp[DIM + 1] = s; }
}

__global__ __launch_bounds__(TPB) void attnpool_combine_kernel(
    const float* __restrict__ partial,
    float*       __restrict__ out)
{
    const int b   = blockIdx.x;
    const int tid = threadIdx.x;
    const float* pb = partial + (size_t)b * CHUNKS * PARTSTRIDE;

    float M = -3.4e38f;
    for (int cix = 0; cix < CHUNKS; ++cix)
        M = fmaxf(M, pb[cix * PARTSTRIDE + DIM]);

    const int d0 = tid * 2;
    v2f acc = {0.0f, 0.0f};
    float S = 0.0f;
    for (int cix = 0; cix < CHUNKS; ++cix) {
        const float* pc = pb + cix * PARTSTRIDE;
        float w = __expf(pc[DIM] - M);
        S += w * pc[DIM + 1];
        v2f a = *(const v2f*)(pc + d0);
        acc += w * a;
    }
    float inv = 1.0f / S;
    out[(size_t)b * DIM + d0]     = acc.x * inv;
    out[(size_t)b * DIM + d0 + 1] = acc.y * inv;
}

extern "C" void kernel_launch(void* const* d_in, const int* in_sizes, int n_in,
                              void* d_out, int out_size, void* d_ws, size_t ws_size,
                              hipStream_t stream) {
    (void)in_sizes; (void)n_in; (void)out_size; (void)ws_size;
    const float* x    = (const float*)d_in[0];
    const int*   mask = (const int*)d_in[1];
    const float* q    = (const float*)d_in[2];
    float* out     = (float*)d_out;
    float* partial = (float*)d_ws;   // BATCH*CHUNKS*PARTSTRIDE floats ~= 4.2 MB

    dim3 grid(CHUNKS, BATCH);
    attnpool_partial_kernel<<<grid, TPB, 0, stream>>>(x, mask, q, partial);
    attnpool_combine_kernel<<<BATCH, TPB, 0, stream>>>(partial, out);
}